// MaxSW_58351425683465
// MI455X (gfx1250) — compile-verified
//
#include <hip/hip_runtime.h>
#include <stdint.h>
#include <cmath>

#define NPTS    131072
#define NBATCH  32
#define NSEG    64          // 32 batches x {x,y}
#define RT      2048        // rows per projection tile
#define TPB_PROJ 256
#define TPB_SORT 1024
#define TPB_RED  1024
#define EPT     8           // radix: elements per thread per chunk
#define CHUNK   (TPB_SORT * EPT)   // 8192
#define ADAM_LR 1e-4f

typedef __attribute__((ext_vector_type(2))) float v2f;
typedef __attribute__((ext_vector_type(8))) float v8f;

__device__ __forceinline__ unsigned f2key(float f) {
  unsigned u = __float_as_uint(f);
  return (u & 0x80000000u) ? ~u : (u | 0x80000000u);
}
__device__ __forceinline__ float key2f(unsigned k) {
  return (k & 0x80000000u) ? __uint_as_float(k ^ 0x80000000u)
                           : __uint_as_float(~k);
}

// ---------------------------------------------------------------------------
// Projection: keys[seg*N + n] = sortable_u32( dot(row_n, p_b) ), idx init.
// seg = 2*b + (0:x, 1:y).  Async-to-LDS staging + WMMA f32 16x16x4.
// A (16x4, documented layout): lane<16: v0=x[r,0](K0), v1=x[r,1](K1);
//                              lane>=16: v0=x[r,2](K2), v1=0(K3).
// B (4x16): every column = p  => v0 = lane<16? p0 : p2 ; v1 = lane<16? p1 : 0.
// D[i][j] = xp_i replicated over columns; 16 lanes pick their row via a
// compile-time select chain -> one coalesced 16-lane store per WMMA.
// ---------------------------------------------------------------------------
__global__ void k_proj(const float* __restrict__ x, const float* __restrict__ y,
                       const float* __restrict__ u,
                       unsigned* __restrict__ keys, unsigned* __restrict__ idx) {
  __shared__ float sm[RT * 3];                       // 24 KB tile
  const int tile = blockIdx.x;                       // 0..63
  const int seg  = blockIdx.y;                       // 0..63
  const int b    = seg >> 1;
  const float* src = (seg & 1) ? y : x;
  const float* rowBase = src + (size_t)b * NPTS * 3 + (size_t)tile * RT * 3;
  const int tid = threadIdx.x;

  // ---- async-tensor path: stage tile into LDS (ASYNCcnt) ----
  const int nVec = RT * 3 / 4;                       // 1536 b128 transfers
  for (int vi = tid; vi < nVec; vi += TPB_PROJ) {
    unsigned lds = (unsigned)(uintptr_t)(&sm[vi * 4]);
    unsigned long long ga = (unsigned long long)(uintptr_t)(rowBase + vi * 4);
    asm volatile("global_load_async_to_lds_b128 %0, %1, off"
                 :: "v"(lds), "v"(ga) : "memory");
  }
  asm volatile("s_wait_asynccnt 0" ::: "memory");
  __syncthreads();

  // ---- p = normalize(u_b), computed redundantly (3 scalar loads) ----
  const float u0 = u[b * 3 + 0], u1 = u[b * 3 + 1], u2 = u[b * 3 + 2];
  const float inv = 1.0f / sqrtf(u0 * u0 + u1 * u1 + u2 * u2);
  const float p0 = u0 * inv, p1 = u1 * inv, p2 = u2 * inv;

  const int lane = tid & 31, wave = tid >> 5;
  const bool lo = lane < 16;
  v2f bm; bm.x = lo ? p0 : p2; bm.y = lo ? p1 : 0.0f;

  const unsigned segBase = (unsigned)seg * NPTS;
  const int rowsPerWave = RT / (TPB_PROJ / 32);      // 256
  const int wrow0 = wave * rowsPerWave;

  for (int g16 = 0; g16 < rowsPerWave; g16 += 16) {
    const int r0  = wrow0 + g16;
    const int row = r0 + (lane & 15);
    // branchless A build: unconditional LDS loads + selects (no EXEC churn)
    v2f a;
    a.x = sm[row * 3 + (lo ? 0 : 2)];
    const float t1 = sm[row * 3 + 1];
    a.y = lo ? t1 : 0.0f;
    v8f c = {};
    v8f d = __builtin_amdgcn_wmma_f32_16x16x4_f32(
        false, a, false, bm, (short)0, c, false, false);
    // D is column-replicated: lane picks d[lane&7]; lanes {0..7,16..23} store.
    float val = d[0];
#pragma unroll
    for (int r = 1; r < 8; ++r) val = ((lane & 7) == r) ? d[r] : val;
    if ((lane & 15) < 8) {
      const int mrow = r0 + (lo ? 0 : 8) + (lane & 7);
      keys[segBase + (unsigned)(tile * RT + mrow)] = f2key(val);
    }
  }
  // idx init, vectorized
  uint4* idxv = (uint4*)(idx + segBase + (unsigned)(tile * RT));
  for (int i = tid; i < RT / 4; i += TPB_PROJ) {
    const unsigned n = (unsigned)(tile * RT + i * 4);
    idxv[i] = make_uint4(n, n + 1u, n + 2u, n + 3u);
  }
}

// ---------------------------------------------------------------------------
// Segmented 4-pass 8-bit LSD radix sort; one workgroup per segment, all
// passes internal. Ranking: 8 elems/thread per 8192-elem chunk in
// wave-blocked round-major order (elem = cbase + wave*256 + j*32 + lane):
//   - in-round lane order  : wave32 ballot digit-match
//   - cross-round order    : per-wave accumulating LDS counters (DS in-order)
//   - cross-wave order     : one 32x256 scan per 8192 elems (8x amortized)
// ---------------------------------------------------------------------------
__global__ void k_sort(unsigned* __restrict__ keysA, unsigned* __restrict__ idxA,
                       unsigned* __restrict__ keysB, unsigned* __restrict__ idxB) {
  __shared__ unsigned hist[256];
  __shared__ unsigned running[256];
  __shared__ unsigned digitTotal[256];
  __shared__ unsigned wcount[32 * 256];              // 32 KB
  const int seg = blockIdx.x;
  const unsigned base = (unsigned)seg * NPTS;
  const int tid = threadIdx.x, lane = tid & 31, wave = tid >> 5;

  unsigned *curK = keysA + base, *curI = idxA + base;
  unsigned *altK = keysB + base, *altI = idxB + base;

  for (int pass = 0; pass < 4; ++pass) {
    const int shift = pass * 8;
    // --- histogram (uint4-vectorized) ---
    if (tid < 256) hist[tid] = 0;
    __syncthreads();
    for (int i = tid; i < NPTS / 4; i += TPB_SORT) {
      const uint4 kk = ((const uint4*)curK)[i];
      atomicAdd(&hist[(kk.x >> shift) & 255u], 1u);
      atomicAdd(&hist[(kk.y >> shift) & 255u], 1u);
      atomicAdd(&hist[(kk.z >> shift) & 255u], 1u);
      atomicAdd(&hist[(kk.w >> shift) & 255u], 1u);
    }
    __syncthreads();
    if (tid == 0) {
      unsigned s = 0;
      for (int d = 0; d < 256; ++d) { unsigned t = hist[d]; running[d] = s; s += t; }
    }
    __syncthreads();
    // --- stable scatter, 8192-element chunks ---
    for (int cbase = 0; cbase < NPTS; cbase += CHUNK) {
      for (int i = tid; i < 32 * 256; i += TPB_SORT) wcount[i] = 0;
      __syncthreads();
      const int wbase = cbase + wave * (32 * EPT);
      unsigned k[EPT], v[EPT], dg[EPT], rk[EPT];
#pragma unroll
      for (int j = 0; j < EPT; ++j) {
        k[j]  = curK[wbase + j * 32 + lane];
        v[j]  = curI[wbase + j * 32 + lane];
        dg[j] = (k[j] >> shift) & 255u;
      }
#pragma unroll
      for (int j = 0; j < EPT; ++j) {
        const unsigned d = dg[j];
        unsigned m = 0xFFFFFFFFu;
#pragma unroll
        for (int bit = 0; bit < 8; ++bit) {
          const unsigned bal = __builtin_amdgcn_ballot_w32(((d >> bit) & 1u) != 0u);
          m &= ((d >> bit) & 1u) ? bal : ~bal;
        }
        const unsigned riw = __popc(m & ((1u << lane) - 1u));
        const unsigned wb  = wcount[wave * 256 + d];   // pre-round count
        rk[j] = wb + riw;
        if (riw == 0) wcount[wave * 256 + d] = wb + __popc(m);  // leader commits
      }
      __syncthreads();
      if (tid < 256) {
        unsigned s = 0;
        for (int w = 0; w < 32; ++w) {
          const unsigned t = wcount[w * 256 + tid];
          wcount[w * 256 + tid] = s; s += t;
        }
        digitTotal[tid] = s;
      }
      __syncthreads();
#pragma unroll
      for (int j = 0; j < EPT; ++j) {
        const unsigned pos = running[dg[j]] + wcount[wave * 256 + dg[j]] + rk[j];
        altK[pos] = k[j]; altI[pos] = v[j];
      }
      __syncthreads();
      if (tid < 256) running[tid] += digitTotal[tid];
      __syncthreads();
    }
    __threadfence();
    __syncthreads();
    unsigned* t;
    t = curK; curK = altK; altK = t;
    t = curI; curI = altI; altI = t;
  }
}

// ---------------------------------------------------------------------------
// grad_p[b] = 2 * sum_i (sxp_i - syp_i) * (x[sx(i)] - y[sy(i)]);  loss_b too.
// One workgroup per batch; fixed-order LDS tree reduction (deterministic).
// ---------------------------------------------------------------------------
__global__ void k_grad(const float* __restrict__ x, const float* __restrict__ y,
                       const unsigned* __restrict__ keys,
                       const unsigned* __restrict__ idx,
                       float* __restrict__ gradp, float* __restrict__ lossb) {
  __shared__ float r0[TPB_RED], r1[TPB_RED], r2[TPB_RED], rL[TPB_RED];
  const int b = blockIdx.x;
  const unsigned bx = (unsigned)(2 * b) * NPTS, by = bx + NPTS;
  const float* xb = x + (size_t)b * NPTS * 3;
  const float* yb = y + (size_t)b * NPTS * 3;
  const int tid = threadIdx.x;
  float a0 = 0.f, a1 = 0.f, a2 = 0.f, aL = 0.f;
  for (int i = tid; i < NPTS; i += TPB_RED) {
    if (i + 8 * TPB_RED < NPTS) {
      __builtin_prefetch(&keys[bx + i + 8 * TPB_RED], 0, 0);
      __builtin_prefetch(&keys[by + i + 8 * TPB_RED], 0, 0);
    }
    const float sx = key2f(keys[bx + i]);
    const float sy = key2f(keys[by + i]);
    const float diff = sx - sy;
    const unsigned ix = idx[bx + i] * 3u, iy = idx[by + i] * 3u;
    a0 += diff * (xb[ix + 0] - yb[iy + 0]);
    a1 += diff * (xb[ix + 1] - yb[iy + 1]);
    a2 += diff * (xb[ix + 2] - yb[iy + 2]);
    aL += diff * diff;
  }
  r0[tid] = a0; r1[tid] = a1; r2[tid] = a2; rL[tid] = aL;
  __syncthreads();
  for (int s = TPB_RED / 2; s > 0; s >>= 1) {
    if (tid < s) {
      r0[tid] += r0[tid + s]; r1[tid] += r1[tid + s];
      r2[tid] += r2[tid + s]; rL[tid] += rL[tid + s];
    }
    __syncthreads();
  }
  if (tid == 0) {
    gradp[b * 3 + 0] = 2.0f * r0[0];
    gradp[b * 3 + 1] = 2.0f * r1[0];
    gradp[b * 3 + 2] = 2.0f * r2[0];
    lossb[b] = rL[0];
  }
}

// Final eval: only sum of squared sorted differences per batch.
__global__ void k_loss(const unsigned* __restrict__ keys, float* __restrict__ lossb) {
  __shared__ float rL[TPB_RED];
  const int b = blockIdx.x;
  const unsigned bx = (unsigned)(2 * b) * NPTS, by = bx + NPTS;
  const int tid = threadIdx.x;
  float aL = 0.f;
  for (int i = tid; i < NPTS; i += TPB_RED) {
    const float diff = key2f(keys[bx + i]) - key2f(keys[by + i]);
    aL += diff * diff;
  }
  rL[tid] = aL;
  __syncthreads();
  for (int s = TPB_RED / 2; s > 0; s >>= 1) {
    if (tid < s) rL[tid] += rL[tid + s];
    __syncthreads();
  }
  if (tid == 0) lossb[b] = rL[0];
}

// Adam step on u (32x3). grad_u = -(1/B) * (I - p p^T)/||u|| * grad_p.
__global__ void k_adam(float* __restrict__ u, float* __restrict__ m,
                       float* __restrict__ v, const float* __restrict__ gradp,
                       float bc1, float bc2) {
  const int b = threadIdx.x;
  if (b >= NBATCH) return;
  const float u0 = u[b*3+0], u1 = u[b*3+1], u2 = u[b*3+2];
  const float nrm = sqrtf(u0*u0 + u1*u1 + u2*u2);
  const float p0 = u0/nrm, p1 = u1/nrm, p2 = u2/nrm;
  const float g0 = gradp[b*3+0], g1 = gradp[b*3+1], g2 = gradp[b*3+2];
  const float pg = p0*g0 + p1*g1 + p2*g2;
  const float s = -1.0f / (32.0f * nrm);
  const float gu0 = s*(g0 - p0*pg), gu1 = s*(g1 - p1*pg), gu2 = s*(g2 - p2*pg);
  float m0 = 0.9f*m[b*3+0] + 0.1f*gu0;
  float m1 = 0.9f*m[b*3+1] + 0.1f*gu1;
  float m2 = 0.9f*m[b*3+2] + 0.1f*gu2;
  float v0 = 0.999f*v[b*3+0] + 0.001f*gu0*gu0;
  float v1 = 0.999f*v[b*3+1] + 0.001f*gu1*gu1;
  float v2 = 0.999f*v[b*3+2] + 0.001f*gu2*gu2;
  m[b*3+0]=m0; m[b*3+1]=m1; m[b*3+2]=m2;
  v[b*3+0]=v0; v[b*3+1]=v1; v[b*3+2]=v2;
  u[b*3+0] = u0 - ADAM_LR * (m0/bc1) / (sqrtf(v0/bc2) + 1e-8f);
  u[b*3+1] = u1 - ADAM_LR * (m1/bc1) / (sqrtf(v1/bc2) + 1e-8f);
  u[b*3+2] = u2 - ADAM_LR * (m2/bc1) / (sqrtf(v2/bc2) + 1e-8f);
}

__global__ void k_init(float* __restrict__ u, float* __restrict__ m,
                       float* __restrict__ v, const float* __restrict__ p0) {
  const int i = threadIdx.x;
  if (i < NBATCH * 3) { u[i] = p0[i]; m[i] = 0.f; v[i] = 0.f; }
}

__global__ void k_out(const float* __restrict__ lossb, float* __restrict__ out) {
  if (threadIdx.x == 0) {
    float s = 0.f;
    for (int i = 0; i < NBATCH; ++i) s += lossb[i];
    out[0] = s / (float)NBATCH;
  }
}

extern "C" void kernel_launch(void* const* d_in, const int* in_sizes, int n_in,
                              void* d_out, int out_size, void* d_ws, size_t ws_size,
                              hipStream_t stream) {
  const float* x  = (const float*)d_in[0];
  const float* y  = (const float*)d_in[1];
  const float* pi = (const float*)d_in[2];

  char* ws = (char*)d_ws;
  float* u     = (float*)(ws + 0);
  float* m     = (float*)(ws + 512);
  float* v     = (float*)(ws + 1024);
  float* gradp = (float*)(ws + 1536);
  float* lossb = (float*)(ws + 2048);
  const size_t SN = (size_t)NSEG * NPTS * sizeof(unsigned);   // 33.55 MB
  unsigned* keysA = (unsigned*)(ws + 4096);
  unsigned* idxA  = (unsigned*)(ws + 4096 + SN);
  unsigned* keysB = (unsigned*)(ws + 4096 + 2 * SN);
  unsigned* idxB  = (unsigned*)(ws + 4096 + 3 * SN);

  k_init<<<1, 128, 0, stream>>>(u, m, v, pi);

  for (int it = 1; it <= 50; ++it) {
    k_proj<<<dim3(NPTS / RT, NSEG), TPB_PROJ, 0, stream>>>(x, y, u, keysA, idxA);
    k_sort<<<NSEG, TPB_SORT, 0, stream>>>(keysA, idxA, keysB, idxB);
    k_grad<<<NBATCH, TPB_RED, 0, stream>>>(x, y, keysA, idxA, gradp, lossb);
    const float bc1 = (float)(1.0 - pow(0.9, (double)it));
    const float bc2 = (float)(1.0 - pow(0.999, (double)it));
    k_adam<<<1, 32, 0, stream>>>(u, m, v, gradp, bc1, bc2);
  }

  k_proj<<<dim3(NPTS / RT, NSEG), TPB_PROJ, 0, stream>>>(x, y, u, keysA, idxA);
  k_sort<<<NSEG, TPB_SORT, 0, stream>>>(keysA, idxA, keysB, idxB);
  k_loss<<<NBATCH, TPB_RED, 0, stream>>>(keysA, lossb);
  k_out<<<1, 32, 0, stream>>>(lossb, (float*)d_out);
}